// KATLayer_6949257085521
// MI455X (gfx1250) — compile-verified
//
#include <hip/hip_runtime.h>
#include <math.h>

// Problem constants (match reference)
constexpr int Bb = 32;   // batch
constexpr int Ii = 512;  // input dim
constexpr int Oo = 512;  // output dim
constexpr int Nn = 16;   // basis count

// Tiling
constexpr int OB     = 8;            // o's per block (one per wave, 8 waves)
constexpr int IB     = 16;           // i-rows per TDM stage
constexpr int ICHUNK = 64;           // i-rows per block
constexpr int STAGES = ICHUNK / IB;  // 4
constexpr int NCHUNK = Ii / ICHUNK;  // 8 partial slabs

typedef unsigned int u32x4 __attribute__((ext_vector_type(4)));
typedef int          i32x4 __attribute__((ext_vector_type(4)));
typedef int          i32x8 __attribute__((ext_vector_type(8)));

// Generic LDS pointer: per the aperture table, addr[31:0] is the LDS byte offset.
#define LDSOFF(p) ((unsigned)(unsigned long long)(const void*)(p))

// Issue one TDM 2-D tile load: IB rows x (OB*Nn) f32 elements, row stride Oo*Nn.
__device__ __forceinline__ void tdm_load_2d_f32(const float* gaddr, unsigned lds_byte_off) {
  unsigned long long ga = (unsigned long long)(const void*)gaddr;
  u32x4 g0;
  g0.x = 1u;                                                  // count=1 (valid user D#)
  g0.y = lds_byte_off;                                        // LDS destination (bytes)
  g0.z = (unsigned)ga;                                        // global_addr[31:0]
  g0.w = (unsigned)((ga >> 32) & 0x01FFFFFFu) | (2u << 30);   // global_addr[56:32] | type=2
  i32x8 g1;
  g1[0] = (int)(2u << 16);                                    // workgroup_mask=0, data_size=2 (4B)
  g1[1] = (int)(((unsigned)(Oo * Nn) & 0xFFFFu) << 16);       // tensor_dim0[15:0]
  g1[2] = (int)((((unsigned)(Oo * Nn) >> 16) & 0xFFFFu)       // tensor_dim0[31:16]
              | (((unsigned)Ii & 0xFFFFu) << 16));            // tensor_dim1[15:0]
  g1[3] = (int)((((unsigned)Ii >> 16) & 0xFFFFu)              // tensor_dim1[31:16]
              | (((unsigned)(OB * Nn) & 0xFFFFu) << 16));     // tile_dim0 = 128
  g1[4] = (int)(IB & 0xFFFF);                                 // tile_dim1 = 16, tile_dim2 = 0
  g1[5] = (int)(Oo * Nn);                                     // tensor_dim0_stride[31:0] = 8192
  g1[6] = 0;                                                  // stride0[47:32] | stride1[15:0]
  g1[7] = 0;                                                  // stride1[47:16]
  i32x4 z4 = {0, 0, 0, 0};
#if __clang_major__ >= 23
  i32x8 z8 = {0, 0, 0, 0, 0, 0, 0, 0};
  __builtin_amdgcn_tensor_load_to_lds(g0, g1, z4, z4, z8, 0);
#else
  __builtin_amdgcn_tensor_load_to_lds(g0, g1, z4, z4, 0);
#endif
}

// x (B x I) -> xT (I x B) so the inner loop reads one coalesced 128B line per i.
__global__ void kat_transpose_x(const float* __restrict__ x, float* __restrict__ xT) {
  __shared__ float t[32][33];
  const int i0 = blockIdx.x * 32;
#pragma unroll
  for (int r = 0; r < 4; ++r) {
    const int b = threadIdx.y + r * 8;
    t[b][threadIdx.x] = x[b * Ii + i0 + threadIdx.x];
  }
  __syncthreads();
#pragma unroll
  for (int r = 0; r < 4; ++r) {
    const int il = threadIdx.y + r * 8;
    xT[(i0 + il) * Bb + threadIdx.x] = t[threadIdx.x][il];
  }
}

// Main kernel: lane = batch b, wave = one output o. TDM double-buffers the
// sigma/alpha/w parameter stream (the only large memory traffic) into LDS.
__global__ __launch_bounds__(256, 4) void kat_forward(
    const float* __restrict__ xT,       const float* __restrict__ mx_train,
    const float* __restrict__ scale,    const float* __restrict__ sigma,
    const float* __restrict__ alpha,    const float* __restrict__ wgt,
    const float* __restrict__ mx_start, float* __restrict__ partial)
{
  __shared__ float sh[2][3][IB * OB * Nn];  // 2 bufs x {sigma,alpha,w} x 2048 f32 = 48KB

  const int wv   = __builtin_amdgcn_readfirstlane((int)(threadIdx.x >> 5));
  const int lane = threadIdx.x & 31;                 // = batch index b
  const int ob0  = blockIdx.x * OB;
  const int o    = ob0 + wv;
  const int ic0  = blockIdx.y * ICHUNK;

  // Prologue: wave 0 kicks off stage 0 (3 TDM descriptors -> TENSORcnt).
  if (wv == 0) {
    const size_t e0 = ((size_t)ic0 * Oo + (size_t)ob0) * Nn;
    tdm_load_2d_f32(sigma + e0, LDSOFF(&sh[0][0][0]));
    tdm_load_2d_f32(alpha + e0, LDSOFF(&sh[0][1][0]));
    tdm_load_2d_f32(wgt   + e0, LDSOFF(&sh[0][2][0]));
  }

  float acc = 0.0f;
#pragma unroll
  for (int s = 0; s < STAGES; ++s) {
    if (wv == 0) {
      if (s + 1 < STAGES) {  // prefetch next stage into the other buffer
        const int nb = (s + 1) & 1;
        const size_t e1 = ((size_t)(ic0 + (s + 1) * IB) * Oo + (size_t)ob0) * Nn;
        tdm_load_2d_f32(sigma + e1, LDSOFF(&sh[nb][0][0]));
        tdm_load_2d_f32(alpha + e1, LDSOFF(&sh[nb][1][0]));
        tdm_load_2d_f32(wgt   + e1, LDSOFF(&sh[nb][2][0]));
        __builtin_amdgcn_s_wait_tensorcnt(3);  // stage s's 3 loads done (in-order)
      } else {
        __builtin_amdgcn_s_wait_tensorcnt(0);
      }
    }
    __syncthreads();  // release all waves: buf[s&1] is valid

    const int buf = s & 1;
    const float* __restrict__ Ssg = sh[buf][0];
    const float* __restrict__ Sal = sh[buf][1];
    const float* __restrict__ Sww = sh[buf][2];

    for (int ii = 0; ii < IB; ++ii) {
      const int   gi = ic0 + s * IB + ii;
      const float xi = xT[gi * Bb + lane];                       // per-lane: x[b, gi]
      const float sc = fabsf(scale[(size_t)gi * Oo + o]);        // wave-uniform -> s_load
      const float mt = mx_train[(size_t)gi * Oo + o];            // wave-uniform -> s_load
      const int   rb = (ii * OB + wv) * Nn;                      // this wave's o-slice
#pragma unroll
      for (int n = 0; n < Nn; ++n) {
        const float sg = Ssg[rb + n];                            // LDS broadcast (uniform)
        const float al = Sal[rb + n];
        const float wn = Sww[rb + n];
        const float c  = fmaf(sc, mx_start[n], mt);              // center
        const float iv = __builtin_amdgcn_rcpf(fabsf(sg) + 1e-8f);
        const float z  = (xi - c) * iv;
        const float e  = __builtin_amdgcn_exp2f(-1.44269504088896340736f * z * z);
        const float t  = erff(0.70710678118654752440f * (al * z));
        acc = fmaf(wn * e, 1.0f + t, acc);
      }
    }
    __syncthreads();  // everyone done with buf[s&1] before it is refilled
  }

  // Deterministic partial write (no float atomics): slab per i-chunk.
  partial[(size_t)blockIdx.y * (Bb * Oo) + (size_t)lane * Oo + o] = acc;
}

// Sum the NCHUNK partial slabs -> out[b, o]; fixed order => deterministic.
__global__ void kat_reduce(const float* __restrict__ partial, float* __restrict__ out) {
  const int k = blockIdx.x * blockDim.x + threadIdx.x;  // 0 .. B*O-1
  float s = 0.0f;
#pragma unroll
  for (int y = 0; y < NCHUNK; ++y) s += partial[(size_t)y * (Bb * Oo) + k];
  out[k] = s;
}

extern "C" void kernel_launch(void* const* d_in, const int* in_sizes, int n_in,
                              void* d_out, int out_size, void* d_ws, size_t ws_size,
                              hipStream_t stream) {
  (void)in_sizes; (void)n_in; (void)out_size; (void)ws_size;
  const float* x        = (const float*)d_in[0];  // (B, I)
  const float* mx_train = (const float*)d_in[1];  // (I, O)
  const float* scale    = (const float*)d_in[2];  // (I, O)
  const float* sigma    = (const float*)d_in[3];  // (I, O, N)
  const float* alpha    = (const float*)d_in[4];  // (I, O, N)
  const float* wgt      = (const float*)d_in[5];  // (I, O, N)
  const float* mx_start = (const float*)d_in[6];  // (N,)
  float* out = (float*)d_out;                     // (B, O)

  float* xT      = (float*)d_ws;                  // I*B floats   (64 KB)
  float* partial = xT + (size_t)Ii * Bb;          // NCHUNK*B*O floats (512 KB)

  kat_transpose_x<<<dim3(Ii / 32), dim3(32, 8), 0, stream>>>(x, xT);
  kat_forward<<<dim3(Oo / OB, Ii / ICHUNK), dim3(256), 0, stream>>>(
      xT, mx_train, scale, sigma, alpha, wgt, mx_start, partial);
  kat_reduce<<<dim3((Bb * Oo) / 256), dim3(256), 0, stream>>>(partial, out);
}